// Decoder_31696858644737
// MI455X (gfx1250) — compile-verified
//
#include <hip/hip_runtime.h>

typedef __attribute__((ext_vector_type(2))) float v2f;
typedef __attribute__((ext_vector_type(8))) float v8f;

// Problem constants (from reference): B=16, S=96, P=96, N=256, D=256
#define BB 16
#define SS 96
#define PP 96
#define NN 256
#define DD 256
// LDS row stride for the attn tile: 98 keeps (row*98 + 4k + 2h) even
// -> 8-byte aligned ds_load_b64 for WMMA A fragments, and only 2-way
// bank conflicts on row-wise softmax scans (98 mod 64 = 34).
#define RS 98

__global__ __launch_bounds__(256) void decoder_attn_kernel(
    const float* __restrict__ enc_time,   // [B,S,N,4]
    const float* __restrict__ enc_tgt,    // [B,N,S,D]
    const float* __restrict__ dec_time,   // [B,P,N,4]
    float* __restrict__ attn_out,         // [B,N,P,S]
    float* __restrict__ tgt_out)          // [B,N,P,D]
{
    __shared__ float4 et4[SS];
    __shared__ float4 dt4[PP];
    __shared__ float  rne[SS];
    __shared__ float  rnd[PP];
    __shared__ float  rsum[PP];
    __shared__ float  attn[PP][RS];

    const int tid = threadIdx.x;
    const int bn  = blockIdx.x;        // b*256 + n
    const int b   = bn >> 8;
    const int n   = bn & 255;

    // ---- stage time vectors (innermost dim == 4 -> float4 loads) + rsqrt norms
    if (tid < SS) {
        float4 e = reinterpret_cast<const float4*>(enc_time)[(b * SS + tid) * NN + n];
        et4[tid] = e;
        rne[tid] = rsqrtf(e.x * e.x + e.y * e.y + e.z * e.z + e.w * e.w);
    } else if (tid >= 128 && tid < 128 + PP) {
        const int p = tid - 128;
        float4 d = reinterpret_cast<const float4*>(dec_time)[(b * PP + p) * NN + n];
        dt4[p] = d;
        rnd[p] = rsqrtf(d.x * d.x + d.y * d.y + d.z * d.z + d.w * d.w);
    }
    __syncthreads();

    // ---- cosine-similarity scores -> LDS (9216 pairs / 256 threads = 36 each)
    for (int i = 0; i < 36; ++i) {
        const int q = tid + 256 * i;
        const int p = q / SS;
        const int s = q - p * SS;
        const float4 dv = dt4[p];
        const float4 ev = et4[s];
        const float sc = dv.x * ev.x + dv.y * ev.y + dv.z * ev.z + dv.w * ev.w;
        attn[p][s] = sc * rnd[p] * rne[s];
    }
    __syncthreads();

    // ---- row softmax: keep UNNORMALIZED exp in LDS, stash 1/sum per row.
    if (tid < PP) {
        float mx = -1e30f;
        for (int s = 0; s < SS; ++s) mx = fmaxf(mx, attn[tid][s]);
        float sum = 0.0f;
        for (int s = 0; s < SS; ++s) {
            const float e = __expf(attn[tid][s] - mx);
            attn[tid][s] = e;
            sum += e;
        }
        rsum[tid] = 1.0f / sum;
    }
    __syncthreads();

    // ---- coalesced global write of normalized attn [P,S] (linear in memory)
    {
        float* aout = attn_out + (size_t)bn * (PP * SS);
        for (int i = 0; i < 36; ++i) {
            const int q = tid + 256 * i;
            const int p = q / SS;
            const int s = q - p * SS;
            aout[q] = attn[p][s] * rsum[p];
        }
    }

    // ---- GEMM: out[p,d] = rsum[p] * sum_s exp[p,s] * tgt[s,d]
    //      M=96 (6 tiles), N=256 (16 tiles), K=96 (24 steps of 4)
    //      via V_WMMA_F32_16X16X4_F32, B fragment reused across 6 M-tiles.
    const float* tgt  = enc_tgt + (size_t)bn * (SS * DD);
    float*       outp = tgt_out + (size_t)bn * (PP * DD);

    const int wave  = tid >> 5;
    const int lane  = tid & 31;
    const int lhalf = lane >> 4;   // 0: K+{0,1} / M=lane ; 1: K+{2,3} / M=lane-16
    const int lcol  = lane & 15;

    for (int nb = wave; nb < 16; nb += 8) {   // uniform trip count -> EXEC all-1s
        const int d0 = nb * 16;
        v8f acc[6] = {};
        for (int k = 0; k < 24; ++k) {
            const int s0 = 4 * k + 2 * lhalf;
            v2f bfrag;
            bfrag.x = tgt[(s0    ) * DD + d0 + lcol];
            bfrag.y = tgt[(s0 + 1) * DD + d0 + lcol];
#pragma unroll
            for (int m = 0; m < 6; ++m) {
                const float* ap = &attn[16 * m + lcol][4 * k + 2 * lhalf];
                v2f afrag;
                afrag.x = ap[0];
                afrag.y = ap[1];
                acc[m] = __builtin_amdgcn_wmma_f32_16x16x4_f32(
                    /*neg_a=*/false, afrag,
                    /*neg_b=*/false, bfrag,
                    /*c_mod=*/(short)0, acc[m],
                    /*reuse_a=*/false, /*reuse_b=*/false);
            }
        }
        // C/D layout: VGPR r -> M = r + 8*lhalf (+16m), N = lcol (+d0)
#pragma unroll
        for (int m = 0; m < 6; ++m) {
#pragma unroll
            for (int r = 0; r < 8; ++r) {
                const int p = 16 * m + r + 8 * lhalf;
                outp[p * DD + d0 + lcol] = acc[m][r] * rsum[p];
            }
        }
    }
}

extern "C" void kernel_launch(void* const* d_in, const int* in_sizes, int n_in,
                              void* d_out, int out_size, void* d_ws, size_t ws_size,
                              hipStream_t stream) {
    (void)in_sizes; (void)n_in; (void)out_size; (void)d_ws; (void)ws_size;
    const float* enc_time = (const float*)d_in[0];  // [B,S,N,4]
    const float* enc_tgt  = (const float*)d_in[1];  // [B,N,S,D]
    const float* dec_time = (const float*)d_in[2];  // [B,P,N,4]

    float* attn_out = (float*)d_out;                               // [B,N,P,S]
    float* tgt_out  = attn_out + (size_t)BB * NN * PP * SS;        // [B,N,P,D]

    decoder_attn_kernel<<<dim3(BB * NN), dim3(256), 0, stream>>>(
        enc_time, enc_tgt, dec_time, attn_out, tgt_out);
}